// GATBlock_2491081031683
// MI455X (gfx1250) — compile-verified
//
#include <hip/hip_runtime.h>
#include <math.h>

typedef __attribute__((ext_vector_type(2))) float v2f;
typedef __attribute__((ext_vector_type(8))) float v8f;

#define NEG_SLOPE 0.2f

__device__ __forceinline__ float lrelu(float v) {
    return v >= 0.f ? v : NEG_SLOPE * v;
}

__device__ __forceinline__ float eluf(float v) {
    return v > 0.f ? v : (__expf(v) - 1.f);
}

// Sign-aware float atomic max (works across mixed signs; init with -inf bits).
__device__ __forceinline__ void atomicMaxF(float* addr, float val) {
    if (val >= 0.f)
        atomicMax((int*)addr, __float_as_int(val));
    else
        atomicMin((unsigned int*)addr, __float_as_uint(val));
}

// ---------------------------------------------------------------------------
// Init: acc1 = b1 broadcast, m1/m2 = -inf, den1/den2 = 0, out = b2 broadcast
// ---------------------------------------------------------------------------
__global__ void init_kernel(float* __restrict__ acc1, float* __restrict__ m1,
                            float* __restrict__ den1, float* __restrict__ m2,
                            float* __restrict__ den2, float* __restrict__ outp,
                            const float* __restrict__ b1, const float* __restrict__ b2,
                            int N) {
    const int i = blockIdx.x * blockDim.x + threadIdx.x;
    const float NEGINF = __int_as_float(0xFF800000);
    if (i < N * 128) acc1[i] = b1[i & 127];
    if (i < N * 32)  outp[i] = b2[i & 31];
    if (i < N * 4)   { m1[i] = NEGINF; den1[i] = 0.f; }
    if (i < N)       { m2[i] = NEGINF; den2[i] = 0.f; }
}

// ---------------------------------------------------------------------------
// C = A(MxK) * B(KxNCOLS), fp32 WMMA 16x16x4, one 16x16 tile per wave32.
// K / NCOLS / ELU are compile-time so the fully-unrolled loop uses immediate
// load offsets (no per-iteration address VALU) and no runtime branches.
// Tiles are enumerated column-major so consecutive waves share the same
// 128x16 B panel (WGP$/L2 reuse) while A rows stream.
// ---------------------------------------------------------------------------
template <int K, int NCOLS, bool ELU>
__global__ void gemm_wmma_f32(const float* __restrict__ A, const float* __restrict__ B,
                              float* __restrict__ C, int M) {
    const int wave = threadIdx.x >> 5;
    const int lane = threadIdx.x & 31;
    constexpr int nTilesN = NCOLS >> 4;
    const int tilesM = (M + 15) >> 4;
    const int tile = blockIdx.x * (blockDim.x >> 5) + wave;
    if (tile >= tilesM * nTilesN) return;          // wave-uniform exit: EXEC all-1s for WMMA
    const int tn = tile / tilesM;                  // column-major tile order
    const int tm = tile - tn * tilesM;
    const int rowBase = tm << 4;
    const int colBase = tn << 4;

    const bool hi = lane >= 16;
    const int half = hi ? lane - 16 : lane;        // A: M index / B,C: N index
    const int koff = hi ? 2 : 0;                   // A/B lane halves carry K={0,1} vs {2,3}

    int arow = rowBase + half;
    if (arow >= M) arow = M - 1;                   // clamp loads; stores are predicated
    const float* __restrict__ Ap = A + (size_t)arow * K + koff;
    const float* __restrict__ Bp = B + (size_t)koff * NCOLS + colBase + half;

    v8f acc = {};
#pragma unroll
    for (int k = 0; k < K; k += 4) {
        float a0 = Ap[k];
        float a1 = Ap[k + 1];
        if (ELU) { a0 = eluf(a0); a1 = eluf(a1); }
        v2f a; a.x = a0; a.y = a1;
        v2f b;
        b.x = Bp[k * NCOLS];
        b.y = Bp[k * NCOLS + NCOLS];
        acc = __builtin_amdgcn_wmma_f32_16x16x4_f32(false, a, false, b,
                                                    (short)0, acc, false, false);
    }

    const int col = colBase + half;
    const int mOff = hi ? 8 : 0;
#pragma unroll
    for (int v = 0; v < 8; ++v) {
        const int r = rowBase + mOff + v;
        if (r < M) C[(size_t)r * NCOLS + col] = acc[v];
    }
}

// ---------------------------------------------------------------------------
// Attention logits, layer 1 (H=4, C=32): wave per node, shuffle reduction.
// ---------------------------------------------------------------------------
__global__ void attn_logits4(const float* __restrict__ xp, const float* __restrict__ a_s,
                             const float* __restrict__ a_d, float* __restrict__ als,
                             float* __restrict__ ald, int N) {
    const int wave = threadIdx.x >> 5;
    const int lane = threadIdx.x & 31;
    const int node = blockIdx.x * (blockDim.x >> 5) + wave;
    if (node >= N) return;
    const float* __restrict__ row = xp + (size_t)node * 128;
#pragma unroll
    for (int h = 0; h < 4; ++h) {
        const float v = row[h * 32 + lane];
        float ps = v * a_s[h * 32 + lane];
        float pd = v * a_d[h * 32 + lane];
#pragma unroll
        for (int off = 16; off > 0; off >>= 1) {
            ps += __shfl_xor(ps, off, 32);
            pd += __shfl_xor(pd, off, 32);
        }
        if (lane == 0) {
            als[node * 4 + h] = ps;
            ald[node * 4 + h] = pd;
        }
    }
}

// Attention logits, layer 2 (H=1, C=32)
__global__ void attn_logits1(const float* __restrict__ xp, const float* __restrict__ a_s,
                             const float* __restrict__ a_d, float* __restrict__ als,
                             float* __restrict__ ald, int N) {
    const int wave = threadIdx.x >> 5;
    const int lane = threadIdx.x & 31;
    const int node = blockIdx.x * (blockDim.x >> 5) + wave;
    if (node >= N) return;
    const float v = xp[(size_t)node * 32 + lane];
    float ps = v * a_s[lane];
    float pd = v * a_d[lane];
#pragma unroll
    for (int off = 16; off > 0; off >>= 1) {
        ps += __shfl_xor(ps, off, 32);
        pd += __shfl_xor(pd, off, 32);
    }
    if (lane == 0) { als[node] = ps; ald[node] = pd; }
}

__device__ __forceinline__ void edge_sd(int t, int E, const int* __restrict__ src,
                                        const int* __restrict__ dst, int& s, int& d) {
    if (t < E) { s = src[t]; d = dst[t]; }
    else       { s = t - E; d = s; }        // appended self-loops
}

// ---------------------------------------------------------------------------
// Layer-1 edge passes (H=4): segment max, segment sum, weighted scatter-add
// ---------------------------------------------------------------------------
__global__ void edge_max4(const int* __restrict__ src, const int* __restrict__ dst,
                          int E, int N, const float4* __restrict__ als,
                          const float4* __restrict__ ald, float* __restrict__ m) {
    const int t = blockIdx.x * blockDim.x + threadIdx.x;
    if (t >= E + N) return;
    int s, d; edge_sd(t, E, src, dst, s, d);
    const float4 as = als[s];
    const float4 ad = ald[d];
    atomicMaxF(&m[d * 4 + 0], lrelu(as.x + ad.x));
    atomicMaxF(&m[d * 4 + 1], lrelu(as.y + ad.y));
    atomicMaxF(&m[d * 4 + 2], lrelu(as.z + ad.z));
    atomicMaxF(&m[d * 4 + 3], lrelu(as.w + ad.w));
}

__global__ void edge_sum4(const int* __restrict__ src, const int* __restrict__ dst,
                          int E, int N, const float4* __restrict__ als,
                          const float4* __restrict__ ald, const float4* __restrict__ m,
                          float* __restrict__ den) {
    const int t = blockIdx.x * blockDim.x + threadIdx.x;
    if (t >= E + N) return;
    int s, d; edge_sd(t, E, src, dst, s, d);
    const float4 as = als[s];
    const float4 ad = ald[d];
    const float4 mm = m[d];
    atomicAdd(&den[d * 4 + 0], __expf(lrelu(as.x + ad.x) - mm.x));
    atomicAdd(&den[d * 4 + 1], __expf(lrelu(as.y + ad.y) - mm.y));
    atomicAdd(&den[d * 4 + 2], __expf(lrelu(as.z + ad.z) - mm.z));
    atomicAdd(&den[d * 4 + 3], __expf(lrelu(as.w + ad.w) - mm.w));
}

__global__ void edge_aggr4(const int* __restrict__ src, const int* __restrict__ dst,
                           int E, int N, const float4* __restrict__ als,
                           const float4* __restrict__ ald, const float4* __restrict__ m,
                           const float4* __restrict__ den, const float* __restrict__ xp,
                           float* __restrict__ acc) {
    const int wave = threadIdx.x >> 5;
    const int lane = threadIdx.x & 31;
    const int t = blockIdx.x * (blockDim.x >> 5) + wave;
    if (t >= E + N) return;
    int s, d; edge_sd(t, E, src, dst, s, d);
    const float4 as = als[s];
    const float4 ad = ald[d];
    const float4 mm = m[d];
    const float4 dd = den[d];
    const float a0 = __expf(lrelu(as.x + ad.x) - mm.x) / (dd.x + 1e-16f);
    const float a1 = __expf(lrelu(as.y + ad.y) - mm.y) / (dd.y + 1e-16f);
    const float a2 = __expf(lrelu(as.z + ad.z) - mm.z) / (dd.z + 1e-16f);
    const float a3 = __expf(lrelu(as.w + ad.w) - mm.w) / (dd.w + 1e-16f);
    const float* __restrict__ xr = xp + (size_t)s * 128;
    float* __restrict__ ar = acc + (size_t)d * 128;
    atomicAdd(ar + lane,      xr[lane]      * a0);
    atomicAdd(ar + 32 + lane, xr[32 + lane] * a1);
    atomicAdd(ar + 64 + lane, xr[64 + lane] * a2);
    atomicAdd(ar + 96 + lane, xr[96 + lane] * a3);
}

// ---------------------------------------------------------------------------
// Layer-2 edge passes (H=1)
// ---------------------------------------------------------------------------
__global__ void edge_max1(const int* __restrict__ src, const int* __restrict__ dst,
                          int E, int N, const float* __restrict__ als,
                          const float* __restrict__ ald, float* __restrict__ m) {
    const int t = blockIdx.x * blockDim.x + threadIdx.x;
    if (t >= E + N) return;
    int s, d; edge_sd(t, E, src, dst, s, d);
    atomicMaxF(&m[d], lrelu(als[s] + ald[d]));
}

__global__ void edge_sum1(const int* __restrict__ src, const int* __restrict__ dst,
                          int E, int N, const float* __restrict__ als,
                          const float* __restrict__ ald, const float* __restrict__ m,
                          float* __restrict__ den) {
    const int t = blockIdx.x * blockDim.x + threadIdx.x;
    if (t >= E + N) return;
    int s, d; edge_sd(t, E, src, dst, s, d);
    atomicAdd(&den[d], __expf(lrelu(als[s] + ald[d]) - m[d]));
}

__global__ void edge_aggr1(const int* __restrict__ src, const int* __restrict__ dst,
                           int E, int N, const float* __restrict__ als,
                           const float* __restrict__ ald, const float* __restrict__ m,
                           const float* __restrict__ den, const float* __restrict__ xp,
                           float* __restrict__ outp) {
    const int wave = threadIdx.x >> 5;
    const int lane = threadIdx.x & 31;
    const int t = blockIdx.x * (blockDim.x >> 5) + wave;
    if (t >= E + N) return;
    int s, d; edge_sd(t, E, src, dst, s, d);
    const float alpha = __expf(lrelu(als[s] + ald[d]) - m[d]) / (den[d] + 1e-16f);
    atomicAdd(&outp[(size_t)d * 32 + lane], xp[(size_t)s * 32 + lane] * alpha);
}

// ---------------------------------------------------------------------------
extern "C" void kernel_launch(void* const* d_in, const int* in_sizes, int n_in,
                              void* d_out, int out_size, void* d_ws, size_t ws_size,
                              hipStream_t stream) {
    const float* x    = (const float*)d_in[0];
    const int*   ei   = (const int*)d_in[1];
    const float* W1   = (const float*)d_in[2];
    const float* a_s1 = (const float*)d_in[3];
    const float* a_d1 = (const float*)d_in[4];
    const float* b1   = (const float*)d_in[5];
    const float* W2   = (const float*)d_in[6];
    const float* a_s2 = (const float*)d_in[7];
    const float* a_d2 = (const float*)d_in[8];
    const float* b2   = (const float*)d_in[9];
    float* out = (float*)d_out;

    const int N = in_sizes[0] / 128;   // F_in = 128
    const int E = in_sizes[1] / 2;     // edge_index is [2, E]
    const int* src = ei;
    const int* dst = ei + E;
    const int Etot = E + N;

    float* ws = (float*)d_ws;
    size_t o = 0;
    float* xp1  = ws + o; o += (size_t)N * 128;
    float* acc1 = ws + o; o += (size_t)N * 128;
    float* als1 = ws + o; o += (size_t)N * 4;
    float* ald1 = ws + o; o += (size_t)N * 4;
    float* m1   = ws + o; o += (size_t)N * 4;
    float* den1 = ws + o; o += (size_t)N * 4;
    float* als2 = ws + o; o += (size_t)N;
    float* ald2 = ws + o; o += (size_t)N;
    float* m2   = ws + o; o += (size_t)N;
    float* den2 = ws + o; o += (size_t)N;
    float* xp2  = xp1;   // xp1 dead after layer-1 aggregation; reuse

    const int THREADS = 256;
    const int WPB = THREADS / 32;

    // 1. init accumulators / softmax state (biases folded in)
    init_kernel<<<(N * 128 + THREADS - 1) / THREADS, THREADS, 0, stream>>>(
        acc1, m1, den1, m2, den2, out, b1, b2, N);

    // 2. xp1 = x @ W1  (WMMA fp32, K=128, 128 output cols)
    {
        const int tiles = ((N + 15) / 16) * (128 / 16);
        gemm_wmma_f32<128, 128, false><<<(tiles + WPB - 1) / WPB, THREADS, 0, stream>>>(
            x, W1, xp1, N);
    }

    // 3. attention logits layer 1
    attn_logits4<<<(N + WPB - 1) / WPB, THREADS, 0, stream>>>(xp1, a_s1, a_d1, als1, ald1, N);

    // 4-6. layer-1 edge softmax + aggregation
    edge_max4<<<(Etot + THREADS - 1) / THREADS, THREADS, 0, stream>>>(
        src, dst, E, N, (const float4*)als1, (const float4*)ald1, m1);
    edge_sum4<<<(Etot + THREADS - 1) / THREADS, THREADS, 0, stream>>>(
        src, dst, E, N, (const float4*)als1, (const float4*)ald1, (const float4*)m1, den1);
    edge_aggr4<<<(Etot + WPB - 1) / WPB, THREADS, 0, stream>>>(
        src, dst, E, N, (const float4*)als1, (const float4*)ald1,
        (const float4*)m1, (const float4*)den1, xp1, acc1);

    // 7. xp2 = elu(acc1) @ W2  (WMMA fp32, ELU fused into A load)
    {
        const int tiles = ((N + 15) / 16) * (32 / 16);
        gemm_wmma_f32<128, 32, true><<<(tiles + WPB - 1) / WPB, THREADS, 0, stream>>>(
            acc1, W2, xp2, N);
    }

    // 8. attention logits layer 2
    attn_logits1<<<(N + WPB - 1) / WPB, THREADS, 0, stream>>>(xp2, a_s2, a_d2, als2, ald2, N);

    // 9-11. layer-2 edge softmax + aggregation into d_out (pre-seeded with b2)
    edge_max1<<<(Etot + THREADS - 1) / THREADS, THREADS, 0, stream>>>(
        src, dst, E, N, als2, ald2, m2);
    edge_sum1<<<(Etot + THREADS - 1) / THREADS, THREADS, 0, stream>>>(
        src, dst, E, N, als2, ald2, m2, den2);
    edge_aggr1<<<(Etot + WPB - 1) / WPB, THREADS, 0, stream>>>(
        src, dst, E, N, als2, ald2, m2, den2, xp2, out);
}